// Homography_87909390614777
// MI455X (gfx1250) — compile-verified
//
#include <hip/hip_runtime.h>

typedef float v2f __attribute__((ext_vector_type(2)));
typedef float v8f __attribute__((ext_vector_type(8)));

#define NB 2
#define TB 8
#define CB 3
#define HB 128
#define WB 128
#define DB 64

// Bilinear sample of 3 channel planes (plane stride 16384 floats) at homogeneous
// point (ncx,ncy,ncz). Channel offsets become load-immediate offsets.
__device__ __forceinline__ void bilinear3(const float* __restrict__ img,
                                          float ncx, float ncy, float ncz,
                                          float& o0, float& o1, float& o2)
{
    float z = ncz;
    if (z == 0.0f) z = 1e-7f;
    const float invz = __builtin_amdgcn_rcpf(z);   // v_rcp_f32, ~1 ulp
    const float gx = ((ncx * invz + 1.0f) * (float)WB - 1.0f) * 0.5f;
    const float gy = ((ncy * invz + 1.0f) * (float)HB - 1.0f) * 0.5f;

    const float x0 = floorf(gx), x1 = x0 + 1.0f;
    const float y0 = floorf(gy), y1 = y0 + 1.0f;
    const float wx1 = gx - x0, wx0 = 1.0f - wx1;
    const float wy1 = gy - y0, wy0 = 1.0f - wy1;

    // clamp (v_med3) needed for the index anyway; validity = (coord == clamp).
    const float xc0 = fminf(fmaxf(x0, 0.0f), (float)(WB - 1));
    const float xc1 = fminf(fmaxf(x1, 0.0f), (float)(WB - 1));
    const float yc0 = fminf(fmaxf(y0, 0.0f), (float)(HB - 1));
    const float yc1 = fminf(fmaxf(y1, 0.0f), (float)(HB - 1));
    // factorized masked weights: corner validity = (x valid) * (y valid)
    const float ux0 = (x0 == xc0) ? wx0 : 0.0f;
    const float ux1 = (x1 == xc1) ? wx1 : 0.0f;
    const float uy0 = (y0 == yc0) ? wy0 : 0.0f;
    const float uy1 = (y1 == yc1) ? wy1 : 0.0f;
    const float w00 = ux0 * uy0, w10 = ux1 * uy0;
    const float w01 = ux0 * uy1, w11 = ux1 * uy1;

    const int ix0 = (int)xc0, ix1 = (int)xc1;
    const int iy0 = (int)yc0, iy1 = (int)yc1;
    const int i00 = (iy0 << 7) | ix0, i10 = (iy0 << 7) | ix1;
    const int i01 = (iy1 << 7) | ix0, i11 = (iy1 << 7) | ix1;

    o0 = w00*img[i00]       + w10*img[i10]       + w01*img[i01]       + w11*img[i11];
    o1 = w00*img[i00+16384] + w10*img[i10+16384] + w01*img[i01+16384] + w11*img[i11+16384];
    o2 = w00*img[i00+32768] + w10*img[i10+32768] + w01*img[i01+32768] + w11*img[i11+32768];
}

// One wave handles (n, t, depth-quad, h); 8 chunks of 16 w's; 4 homographies per
// WMMA: rows 0-2,3-5 -> depths d0,d1 (owned by lanes 0-15 via D VGPRs 0-5),
// rows 8-10,11-13 -> depths d2,d3 (owned by lanes 16-31).
// unit bits: h[6:0] | dq[10:7] | t[13:11] | n[14]  -> 32768 units, 8/block.
__global__ __launch_bounds__(256) void homog_warp_kernel(
    const float* __restrict__ images,
    const float* __restrict__ intr,
    const float* __restrict__ extr,
    const float* __restrict__ depths,
    float* __restrict__ out)
{
    const int lane = threadIdx.x & 31;
    const int wave = threadIdx.x >> 5;
    const int unit = blockIdx.x * 8 + wave;

    const int h  = unit & 127;
    const int dq = (unit >> 7) & 15;
    const int t  = (unit >> 11) & 7;
    const int n  = (unit >> 14) & 1;

    // ---- uniform per-(n,t) setup ----
    const float* Kr = intr + (size_t)(n * TB + 0) * 9;
    const float* Er = extr + (size_t)(n * TB + 0) * 16;
    const float* Kv = intr + (size_t)(n * TB + t) * 9;
    const float* Ev = extr + (size_t)(n * TB + t) * 16;

    const float k00 = Kr[0], k01 = Kr[1], k02 = Kr[2];
    const float k10 = Kr[3], k11 = Kr[4], k12 = Kr[5];
    const float k20 = Kr[6], k21 = Kr[7], k22 = Kr[8];
    const float c00 = k11 * k22 - k12 * k21;
    const float c01 = k02 * k21 - k01 * k22;
    const float c02 = k01 * k12 - k02 * k11;
    const float c10 = k12 * k20 - k10 * k22;
    const float c11 = k00 * k22 - k02 * k20;
    const float c12 = k02 * k10 - k00 * k12;
    const float c20 = k10 * k21 - k11 * k20;
    const float c21 = k01 * k20 - k00 * k21;
    const float c22 = k00 * k11 - k01 * k10;
    const float invdet = 1.0f / (k00 * c00 + k01 * c10 + k02 * c20);
    const float Ki00 = c00 * invdet, Ki01 = c01 * invdet, Ki02 = c02 * invdet;
    const float Ki10 = c10 * invdet, Ki11 = c11 * invdet, Ki12 = c12 * invdet;
    const float Ki20 = c20 * invdet, Ki21 = c21 * invdet, Ki22 = c22 * invdet;

    const float Rr00 = Er[0],  Rr01 = Er[1],  Rr02 = Er[2],  tr0 = Er[3];
    const float Rr10 = Er[4],  Rr11 = Er[5],  Rr12 = Er[6],  tr1 = Er[7];
    const float Rr20 = Er[8],  Rr21 = Er[9],  Rr22 = Er[10], tr2 = Er[11];
    const float nr0 = Rr20, nr1 = Rr21, nr2 = Rr22;

    const float rM00 = Rr00*Ki00 + Rr01*Ki10 + Rr02*Ki20;
    const float rM01 = Rr00*Ki01 + Rr01*Ki11 + Rr02*Ki21;
    const float rM02 = Rr00*Ki02 + Rr01*Ki12 + Rr02*Ki22;
    const float rM10 = Rr10*Ki00 + Rr11*Ki10 + Rr12*Ki20;
    const float rM11 = Rr10*Ki01 + Rr11*Ki11 + Rr12*Ki21;
    const float rM12 = Rr10*Ki02 + Rr11*Ki12 + Rr12*Ki22;
    const float rM20 = Rr20*Ki00 + Rr21*Ki10 + Rr22*Ki20;
    const float rM21 = Rr20*Ki01 + Rr21*Ki11 + Rr22*Ki21;
    const float rM22 = Rr20*Ki02 + Rr21*Ki12 + Rr22*Ki22;

    const float a0 = Rr00*tr0 + Rr10*tr1 + Rr20*tr2;
    const float a1 = Rr01*tr0 + Rr11*tr1 + Rr21*tr2;
    const float a2 = Rr02*tr0 + Rr12*tr1 + Rr22*tr2;

    const float Rv00 = Ev[0],  Rv01 = Ev[1],  Rv02 = Ev[2],  tv0 = Ev[3];
    const float Rv10 = Ev[4],  Rv11 = Ev[5],  Rv12 = Ev[6],  tv1 = Ev[7];
    const float Rv20 = Ev[8],  Rv21 = Ev[9],  Rv22 = Ev[10], tv2 = Ev[11];

    const float d0v = a0 - (Rv00*tv0 + Rv10*tv1 + Rv20*tv2);
    const float d1v = a1 - (Rv01*tv0 + Rv11*tv1 + Rv21*tv2);
    const float d2v = a2 - (Rv02*tv0 + Rv12*tv1 + Rv22*tv2);

    const float KR00 = Kv[0]*Rv00 + Kv[1]*Rv10 + Kv[2]*Rv20;
    const float KR01 = Kv[0]*Rv01 + Kv[1]*Rv11 + Kv[2]*Rv21;
    const float KR02 = Kv[0]*Rv02 + Kv[1]*Rv12 + Kv[2]*Rv22;
    const float KR10 = Kv[3]*Rv00 + Kv[4]*Rv10 + Kv[5]*Rv20;
    const float KR11 = Kv[3]*Rv01 + Kv[4]*Rv11 + Kv[5]*Rv21;
    const float KR12 = Kv[3]*Rv02 + Kv[4]*Rv12 + Kv[5]*Rv22;
    const float KR20 = Kv[6]*Rv00 + Kv[7]*Rv10 + Kv[8]*Rv20;
    const float KR21 = Kv[6]*Rv01 + Kv[7]*Rv11 + Kv[8]*Rv21;
    const float KR22 = Kv[6]*Rv02 + Kv[7]*Rv12 + Kv[8]*Rv22;

    const float kd0 = KR00*d0v + KR01*d1v + KR02*d2v;
    const float kd1 = KR10*d0v + KR11*d1v + KR12*d2v;
    const float kd2 = KR20*d0v + KR21*d1v + KR22*d2v;

    const float G00 = KR00*rM00 + KR01*rM10 + KR02*rM20;
    const float G01 = KR00*rM01 + KR01*rM11 + KR02*rM21;
    const float G02 = KR00*rM02 + KR01*rM12 + KR02*rM22;
    const float G10 = KR10*rM00 + KR11*rM10 + KR12*rM20;
    const float G11 = KR10*rM01 + KR11*rM11 + KR12*rM21;
    const float G12 = KR10*rM02 + KR11*rM12 + KR12*rM22;
    const float G20 = KR20*rM00 + KR21*rM10 + KR22*rM20;
    const float G21 = KR20*rM01 + KR21*rM11 + KR22*rM21;
    const float G22 = KR20*rM02 + KR21*rM12 + KR22*rM22;

    const float m0 = nr0*rM00 + nr1*rM10 + nr2*rM20;
    const float m1 = nr0*rM01 + nr1*rM11 + nr2*rM21;
    const float m2 = nr0*rM02 + nr1*rM12 + nr2*rM22;

    // ---- per-lane A fragment: 4 homographies in rows 0-2,3-5,8-10,11-13 ----
    const int  base_d = dq * 4;
    const float dv0 = depths[base_d + 0];
    const float dv1 = depths[base_d + 1];
    const float dv2 = depths[base_d + 2];
    const float dv3 = depths[base_d + 3];

    const int  Mrow = lane & 15;
    const bool hihalf = lane >= 16;
    const int  mm = Mrow & 7;            // row within half: 0-5 valid, 6-7 zero
    const bool vrow = (mm < 6);
    const bool second = (mm >= 3);
    const int  r = second ? (mm - 3) : mm;

    const float dep = (Mrow < 8) ? (second ? dv1 : dv0)
                                 : (second ? dv3 : dv2);
    float sr0, sr1, sr2, skd;
    sr0 = !vrow ? 0.0f : ((r == 0) ? G00 : ((r == 1) ? G10 : G20));
    sr1 = !vrow ? 0.0f : ((r == 0) ? G01 : ((r == 1) ? G11 : G21));
    sr2 = !vrow ? 0.0f : ((r == 0) ? G02 : ((r == 1) ? G12 : G22));
    skd = !vrow ? 0.0f : ((r == 0) ? kd0 : ((r == 1) ? kd1 : kd2));
    const float s = skd / dep;           // depths >= 1.0, once per wave
    const float row0 = sr0 - s * m0;
    const float row1 = sr1 - s * m1;
    const float row2 = sr2 - s * m2;

    // A (16x4 f32): lanes 0-15 hold K0/K1 of row M, lanes 16-31 hold K2/K3.
    v2f Af;
    Af.x = hihalf ? row2 : row0;
    Af.y = hihalf ? 0.0f : row1;         // K3 column is all-zero

    // B (4x16 f32): VGPR0 = row0 (x) / row2 (ones); VGPR1 = row1 (w) / row3 (don't
    // care, since A's K3 column is zero) -> by is a plain float += 16 per chunk.
    const int  ncol = lane & 15;
    const float bx = hihalf ? 1.0f : ((float)h + 0.5f);
    float by = (float)ncol + 0.5f;

    const int   dA = base_d + (hihalf ? 2 : 0);   // this lane's first depth
    const int   nt = n * TB + t;
    const float* imgbase = images + (size_t)nt * CB * HB * WB;
    float* outc0 = out + ((((size_t)nt * CB + 0) * DB + dA) * HB + h) * WB;
    float* outc1 = out + ((((size_t)nt * CB + 1) * DB + dA) * HB + h) * WB;
    float* outc2 = out + ((((size_t)nt * CB + 2) * DB + dA) * HB + h) * WB;

    #pragma unroll 2
    for (int w16 = 0; w16 < 8; ++w16) {
        v2f Bf;
        Bf.x = bx;
        Bf.y = by;
        by += 16.0f;

        v8f Cz = {0.0f, 0.0f, 0.0f, 0.0f, 0.0f, 0.0f, 0.0f, 0.0f};
        v8f Dd = __builtin_amdgcn_wmma_f32_16x16x4_f32(
            false, Af, false, Bf, (short)0, Cz, false, false);

        // lane owns two points: depth dA (Dd[0..2]) and dA+1 (Dd[3..5])
        float a0o, a1o, a2o, b0o, b1o, b2o;
        bilinear3(imgbase, Dd[0], Dd[1], Dd[2], a0o, a1o, a2o);
        bilinear3(imgbase, Dd[3], Dd[4], Dd[5], b0o, b1o, b2o);

        const int wi = w16 * 16 + ncol;
        // depth dA+1 plane is +HB*WB floats -> folds into store imm offset
        __builtin_nontemporal_store(a0o, outc0 + wi);
        __builtin_nontemporal_store(b0o, outc0 + wi + HB * WB);
        __builtin_nontemporal_store(a1o, outc1 + wi);
        __builtin_nontemporal_store(b1o, outc1 + wi + HB * WB);
        __builtin_nontemporal_store(a2o, outc2 + wi);
        __builtin_nontemporal_store(b2o, outc2 + wi + HB * WB);
    }
}

extern "C" void kernel_launch(void* const* d_in, const int* in_sizes, int n_in,
                              void* d_out, int out_size, void* d_ws, size_t ws_size,
                              hipStream_t stream) {
    (void)in_sizes; (void)n_in; (void)out_size; (void)d_ws; (void)ws_size;
    const float* images = (const float*)d_in[0];
    const float* intrin = (const float*)d_in[1];
    const float* extrin = (const float*)d_in[2];
    const float* depths = (const float*)d_in[3];
    float* outp = (float*)d_out;

    // 32768 wave-units (n*t*dq*h = 2*8*16*128), 8 waves per 256-thread block.
    dim3 grid(4096), block(256);
    homog_warp_kernel<<<grid, block, 0, stream>>>(images, intrin, extrin, depths, outp);
}